// NeuralGeodesicFlows_83854941487897
// MI455X (gfx1250) — compile-verified
//
#include <hip/hip_runtime.h>
#include <math.h>

typedef float v2f __attribute__((ext_vector_type(2)));
typedef float v8f __attribute__((ext_vector_type(8)));

#define MDIM   8
#define HID    64
#define NWAVES 4
#define ST     65   // staging tile row stride (floats), padded vs 64
#define W1S    65   // LDS W1 row stride
#define W2S    65   // LDS W2 row stride
#define ZSS    17   // z state row stride (16 + pad)

// fp32 WMMA: D(16x16,f32) = A(16x4,f32) * B(4x16,f32) + C
// 8-arg pattern: (neg_a, A, neg_b, B, c_mod, C, reuse_a, reuse_b)
__device__ __forceinline__ v8f wmma4(v2f a, v2f b, v8f c) {
  return __builtin_amdgcn_wmma_f32_16x16x4_f32(false, a, false, b, (short)0, c,
                                               false, false);
}

// Branch-free tanh: tanh(x) = 1 - 2/(exp(2x)+1).
// __expf -> v_exp_f32; __builtin_amdgcn_rcpf -> raw v_rcp_f32 (no IEEE-div
// Newton sequence). Saturation handled naturally: exp->inf gives 1,
// exp->0 gives -1.
__device__ __forceinline__ float fast_tanh(float x) {
  float e = __expf(2.f * x);
  return 1.f - 2.f * __builtin_amdgcn_rcpf(e + 1.f);
}

__global__ __launch_bounds__(128)
void NeuralGeodesicFlows_83854941487897_kernel(
    const float* __restrict__ z, const float* __restrict__ W1,
    const float* __restrict__ b1, const float* __restrict__ W2,
    const float* __restrict__ b2, const float* __restrict__ tptr,
    const int* __restrict__ nstepsp, float* __restrict__ out) {
  __shared__ float sW1[MDIM * W1S];
  __shared__ float sW2[HID * W2S];
  __shared__ float sb1[HID];
  __shared__ float sb2[HID];
  __shared__ float stageAll[NWAVES][16 * ST];
  __shared__ float LbufAll[NWAVES][16 * 36];
  __shared__ float zcAll[NWAVES][16 * ZSS];
  __shared__ float zsAll[NWAVES][16 * ZSS];

  const int tid  = threadIdx.x;
  const int wave = tid >> 5;
  const int lane = tid & 31;
  const int n16  = lane & 15;  // column / point-row index within 16
  const int hi   = lane >> 4;  // lane half (selects K pair / M half)

  float* stage = stageAll[wave];
  float* Lbuf  = LbufAll[wave];
  float* zc    = zcAll[wave];
  float* zs    = zsAll[wave];

  // ---- cooperative preload of weights into LDS (padded rows) ----
  for (int i = tid; i < MDIM * HID; i += blockDim.x)
    sW1[(i >> 6) * W1S + (i & 63)] = W1[i];
  for (int i = tid; i < HID * HID; i += blockDim.x)
    sW2[(i >> 6) * W2S + (i & 63)] = W2[i];
  if (tid < HID) { sb1[tid] = b1[tid]; sb2[tid] = b2[tid]; }

  const int tile0 = blockIdx.x * (NWAVES * 16) + wave * 16;  // first point of tile
  for (int i = lane; i < 16 * 16; i += 32)
    zc[(i >> 4) * ZSS + (i & 15)] = z[(tile0 + (i >> 4)) * 16 + (i & 15)];

  const int nsteps = nstepsp[0];
  const float dt = tptr[0] / (float)nsteps;
  __syncthreads();

  float kacc[16], kcur[16];

  for (int step = 0; step < nsteps; ++step) {
    for (int stg = 0; stg < 4; ++stg) {
      // ---- build RK4 stage input zs = zc + coef*dt*k_prev ----
      if (lane < 16) {
        const float coef = (stg == 3) ? dt : 0.5f * dt;
        #pragma unroll
        for (int j = 0; j < 16; ++j)
          zs[lane * ZSS + j] =
              zc[lane * ZSS + j] + ((stg == 0) ? 0.f : coef * kcur[j]);
      }
      __syncthreads();

      // =================== RHS evaluation ===================
      // Step A/B: U = X @ W1 + b1 (WMMA, K=8 as 2 chunks of 4);
      //           H = tanh(U); keep h2 = 1 - H^2; H tile -> stage
      v8f h2v[4];
      #pragma unroll
      for (int nb = 0; nb < 4; ++nb) {
        v8f c;
        const float bias = sb1[nb * 16 + n16];
        #pragma unroll
        for (int r = 0; r < 8; ++r) c[r] = bias;
        #pragma unroll
        for (int kc = 0; kc < 2; ++kc) {
          const int k0 = 4 * kc + 2 * hi;
          v2f a, b;
          a.x = zs[n16 * ZSS + k0];       // X[m][k], m = lane%16
          a.y = zs[n16 * ZSS + k0 + 1];
          b.x = sW1[k0 * W1S + nb * 16 + n16];        // W1[k][n]
          b.y = sW1[(k0 + 1) * W1S + nb * 16 + n16];
          c = wmma4(a, b, c);
        }
        #pragma unroll
        for (int r = 0; r < 8; ++r) {
          float h = fast_tanh(c[r]);
          h2v[nb][r] = 1.f - h * h;
          stage[(r + 8 * hi) * ST + nb * 16 + n16] = h;  // C-layout -> tile
        }
      }
      __syncthreads();

      // Step C: Amat = H @ W2 + b2  (16x64 tile; 64 WMMAs)
      v8f am[4];
      #pragma unroll
      for (int nb = 0; nb < 4; ++nb) {
        const float bias = sb2[nb * 16 + n16];
        #pragma unroll
        for (int r = 0; r < 8; ++r) am[nb][r] = bias;
      }
      #pragma unroll
      for (int kc = 0; kc < 16; ++kc) {
        const int k0 = 4 * kc + 2 * hi;
        v2f a;
        a.x = stage[n16 * ST + k0];   // H[m][k]
        a.y = stage[n16 * ST + k0 + 1];
        #pragma unroll
        for (int nb = 0; nb < 4; ++nb) {
          v2f b;
          b.x = sW2[k0 * W2S + nb * 16 + n16];        // W2[k][n]
          b.y = sW2[(k0 + 1) * W2S + nb * 16 + n16];
          am[nb] = wmma4(a, b, am[nb]);
        }
      }
      __syncthreads();
      #pragma unroll
      for (int nb = 0; nb < 4; ++nb)
        #pragma unroll
        for (int r = 0; r < 8; ++r)
          stage[(r + 8 * hi) * ST + nb * 16 + n16] = am[nb][r];  // Amat tile
      __syncthreads();

      // Step D (per point, lane p): g = A A^T + I, Cholesky g = L L^T
      //   (diagonal stored as reciprocal -> solves become multiplies),
      //   vp = A^T v, outer = v (x) vp -> overwrite own stage row
      if (lane < 16) {
        const float* Ar = stage + lane * ST;  // A[a][q] = Ar[a*8+q]
        float Lm[8][8];                       // diag holds 1/L[k][k]
        {
          float g[8][8];
          #pragma unroll
          for (int a = 0; a < 8; ++a)
            #pragma unroll
            for (int b = 0; b < 8; ++b) {
              if (b > a) continue;
              float s = (a == b) ? 1.f : 0.f;
              #pragma unroll
              for (int q = 0; q < 8; ++q) s += Ar[a * 8 + q] * Ar[b * 8 + q];
              g[a][b] = s;
            }
          #pragma unroll
          for (int k = 0; k < 8; ++k) {
            float s = g[k][k];
            #pragma unroll
            for (int j = 0; j < 8; ++j)
              if (j < k) s -= Lm[k][j] * Lm[k][j];
            const float d = sqrtf(fmaxf(s, 1e-20f));
            const float inv = __builtin_amdgcn_rcpf(d);  // v_rcp_f32
            Lm[k][k] = inv;                              // store reciprocal
            #pragma unroll
            for (int r = 0; r < 8; ++r)
              if (r > k) {
                float s2 = g[r][k];
                #pragma unroll
                for (int j = 0; j < 8; ++j)
                  if (j < k) s2 -= Lm[r][j] * Lm[k][j];
                Lm[r][k] = s2 * inv;
              }
          }
        }
        {  // stash lower triangle (36) for the final solve
          int idx = 0;
          #pragma unroll
          for (int k = 0; k < 8; ++k)
            #pragma unroll
            for (int r = 0; r < 8; ++r)
              if (r >= k) Lbuf[lane * 36 + (idx++)] = Lm[r][k];
        }
        float vv[8], vp[8];
        #pragma unroll
        for (int i = 0; i < 8; ++i) vv[i] = zs[lane * ZSS + 8 + i];
        #pragma unroll
        for (int q = 0; q < 8; ++q) {
          float s = 0.f;
          #pragma unroll
          for (int a = 0; a < 8; ++a) s += vv[a] * Ar[a * 8 + q];
          vp[q] = s;
        }
        #pragma unroll
        for (int p = 0; p < 8; ++p)
          #pragma unroll
          for (int q = 0; q < 8; ++q)
            stage[lane * ST + p * 8 + q] = vv[p] * vp[q];  // outer row
      }
      __syncthreads();

      // Step E: S = OUTER @ W2^T  (transposed B read; 64 WMMAs)
      v8f sv[4];
      #pragma unroll
      for (int nb = 0; nb < 4; ++nb)
        #pragma unroll
        for (int r = 0; r < 8; ++r) sv[nb][r] = 0.f;
      #pragma unroll
      for (int kc = 0; kc < 16; ++kc) {
        const int k0 = 4 * kc + 2 * hi;
        v2f a;
        a.x = stage[n16 * ST + k0];
        a.y = stage[n16 * ST + k0 + 1];
        #pragma unroll
        for (int nb = 0; nb < 4; ++nb) {
          v2f b;  // B[k][j] = W2[j][k]
          b.x = sW2[(nb * 16 + n16) * W2S + k0];
          b.y = sW2[(nb * 16 + n16) * W2S + k0 + 1];
          sv[nb] = wmma4(a, b, sv[nb]);
        }
      }
      __syncthreads();
      // Step F: W = h2 (.) S -> stage (layouts match)
      #pragma unroll
      for (int nb = 0; nb < 4; ++nb)
        #pragma unroll
        for (int r = 0; r < 8; ++r)
          stage[(r + 8 * hi) * ST + nb * 16 + n16] = h2v[nb][r] * sv[nb][r];
      __syncthreads();

      // Step G: ct = W @ W1^T  (N=8 padded to 16; 16 WMMAs)
      v8f cvv;
      #pragma unroll
      for (int r = 0; r < 8; ++r) cvv[r] = 0.f;
      {
        const bool valid = (n16 < 8);
        #pragma unroll
        for (int kc = 0; kc < 16; ++kc) {
          const int k0 = 4 * kc + 2 * hi;
          v2f a, b;
          a.x = stage[n16 * ST + k0];
          a.y = stage[n16 * ST + k0 + 1];
          b.x = valid ? sW1[n16 * W1S + k0] : 0.f;      // W1^T[k][i] = W1[i][k]
          b.y = valid ? sW1[n16 * W1S + k0 + 1] : 0.f;
          cvv = wmma4(a, b, cvv);
        }
      }
      __syncthreads();
      #pragma unroll
      for (int r = 0; r < 8; ++r)
        stage[(r + 8 * hi) * ST + n16] = cvv[r];
      __syncthreads();

      // Step H: dv = -g^{-1} ct via stored Cholesky (reciprocal diag ->
      // triangular solves are multiply-only)
      float dv[8];
      if (lane < 16) {
        float ct[8], y[8], zt[8];
        #pragma unroll
        for (int i = 0; i < 8; ++i) ct[i] = stage[lane * ST + i];
        float Lm[8][8];
        {
          int idx = 0;
          #pragma unroll
          for (int k = 0; k < 8; ++k)
            #pragma unroll
            for (int r = 0; r < 8; ++r)
              if (r >= k) Lm[r][k] = Lbuf[lane * 36 + (idx++)];
        }
        #pragma unroll
        for (int r = 0; r < 8; ++r) {  // L y = ct
          float s = ct[r];
          #pragma unroll
          for (int k = 0; k < 8; ++k)
            if (k < r) s -= Lm[r][k] * y[k];
          y[r] = s * Lm[r][r];         // diag is reciprocal
        }
        #pragma unroll
        for (int r = 7; r >= 0; --r) {  // L^T zt = y
          float s = y[r];
          #pragma unroll
          for (int k = 0; k < 8; ++k)
            if (k > r) s -= Lm[k][r] * zt[k];
          zt[r] = s * Lm[r][r];        // diag is reciprocal
        }
        #pragma unroll
        for (int i = 0; i < 8; ++i) dv[i] = -zt[i];
      }
      // =================== end RHS ===================

      if (lane < 16) {
        #pragma unroll
        for (int j = 0; j < 8; ++j) kcur[j] = zs[lane * ZSS + 8 + j];  // dx = v
        #pragma unroll
        for (int j = 0; j < 8; ++j) kcur[8 + j] = dv[j];
        if (stg == 0) {
          #pragma unroll
          for (int j = 0; j < 16; ++j) kacc[j] = kcur[j];
        } else {
          const float w = (stg == 3) ? 1.f : 2.f;
          #pragma unroll
          for (int j = 0; j < 16; ++j) kacc[j] += w * kcur[j];
        }
      }
      __syncthreads();
    }  // stg

    if (lane < 16) {
      #pragma unroll
      for (int j = 0; j < 16; ++j)
        zc[lane * ZSS + j] += (dt / 6.f) * kacc[j];
    }
    __syncthreads();
  }  // step

  for (int i = lane; i < 16 * 16; i += 32)
    out[(tile0 + (i >> 4)) * 16 + (i & 15)] = zc[(i >> 4) * ZSS + (i & 15)];
}

extern "C" void kernel_launch(void* const* d_in, const int* in_sizes, int n_in,
                              void* d_out, int out_size, void* d_ws,
                              size_t ws_size, hipStream_t stream) {
  (void)n_in; (void)out_size; (void)d_ws; (void)ws_size;
  const float* z  = (const float*)d_in[0];
  const float* W1 = (const float*)d_in[1];
  const float* b1 = (const float*)d_in[2];
  const float* W2 = (const float*)d_in[3];
  const float* b2 = (const float*)d_in[4];
  const float* t  = (const float*)d_in[5];
  const int*  ns  = (const int*)d_in[6];
  float* out      = (float*)d_out;

  const int npts = in_sizes[0] / 16;            // B = 16384
  const int grid = npts / (NWAVES * 16);        // 256 blocks, 4 waves each
  hipLaunchKernelGGL(NeuralGeodesicFlows_83854941487897_kernel, dim3(grid),
                     dim3(NWAVES * 32), 0, stream, z, W1, b1, W2, b2, t, ns,
                     out);
}